// BAHDANAUplus_48971217109154
// MI455X (gfx1250) — compile-verified
//
#include <hip/hip_runtime.h>
#include <hip/hip_bf16.h>
#include <math.h>

typedef __attribute__((ext_vector_type(2))) float v2f;
typedef __attribute__((ext_vector_type(8))) float v8f;

#define WAVES 4
#define TILE  16

// padded strides chosen so the 16 per-lane weight rows map to distinct LDS banks
#define AW_STRIDE 132   // (132 % 64)*n distinct for n=0..15
#define W1_STRIDE 100   // (100 % 64)*n distinct for n=0..15

__launch_bounds__(128, 1)
__global__ void bahdanau_fused_kernel(
    const int* __restrict__ group_inputs,
    const int* __restrict__ item_inputs,
    const int* __restrict__ members,
    const float* __restrict__ user_emb,
    const float* __restrict__ item_emb,
    const float* __restrict__ genres,
    const float* __restrict__ attn_W,
    const float* __restrict__ attn_b,
    const float* __restrict__ pred_W1,
    const float* __restrict__ pred_b1,
    const float* __restrict__ pred_W2,
    const float* __restrict__ pred_b2,
    float* __restrict__ out,
    int B)
{
    __shared__ float sh_gi[WAVES][TILE][128];     // member rows + item row (later: "new")
    __shared__ float sh_at[WAVES][TILE][4];       // attention logits (+bias)
    __shared__ float sh_h [WAVES][TILE][8];       // relu hidden
    __shared__ float sh_aWT[16][AW_STRIDE];       // attn_W zero-padded, transposed [n][k]
    __shared__ float sh_W1T[16][W1_STRIDE];       // pred_W1 zero-padded, transposed [n][k]
    __shared__ float sh_attnb[3];
    __shared__ float sh_b1[8];
    __shared__ float sh_W2[8];
    __shared__ float sh_b2;

    const int tid  = threadIdx.x;
    const int wave = tid >> 5;
    const int lane = tid & 31;

    // ---- stage weights into LDS in B-fragment order, zero padded (once per block) ----
    for (int i = tid; i < 16 * AW_STRIDE; i += blockDim.x) {
        const int nn = i / AW_STRIDE;
        const int k  = i - nn * AW_STRIDE;
        sh_aWT[nn][k] = (nn < 3 && k < 128) ? attn_W[k * 3 + nn] : 0.0f;
    }
    for (int i = tid; i < 16 * W1_STRIDE; i += blockDim.x) {
        const int nn = i / W1_STRIDE;
        const int k  = i - nn * W1_STRIDE;
        sh_W1T[nn][k] = (nn < 8 && k < 96) ? pred_W1[k * 8 + nn] : 0.0f;
    }
    if (tid < 3) sh_attnb[tid] = attn_b[tid];
    if (tid < 8) { sh_b1[tid] = pred_b1[tid]; sh_W2[tid] = pred_W2[tid]; }
    if (tid == 0) sh_b2 = pred_b2[0];

    const int s0 = (blockIdx.x * WAVES + wave) * TILE;
    float (*gi)[128] = sh_gi[wave];

    // ---- gather: 64 rows (16 samples x [3 members + 1 item]) over 32 lanes ----
    #pragma unroll
    for (int rr = 0; rr < 2; ++rr) {
        const int r    = lane + rr * 32;   // 0..63
        const int m    = r >> 2;           // sample within tile
        const int part = r & 3;            // 0..2 member, 3 item
        int gs = s0 + m; if (gs >= B) gs = 0;   // clamp (keeps waves uniform)
        if (part < 3) {
            const int grp = group_inputs[gs];
            const int u   = members[(size_t)grp * 3 + part];
            const float4* src = reinterpret_cast<const float4*>(user_emb) + (size_t)u * 8;
            float* dst = &gi[m][part * 32];
            #pragma unroll
            for (int j = 0; j < 8; ++j) {
                float4 v = src[j];
                dst[4 * j + 0] = v.x; dst[4 * j + 1] = v.y;
                dst[4 * j + 2] = v.z; dst[4 * j + 3] = v.w;
            }
        } else {
            const int it = item_inputs[gs];
            const float* ie = item_emb + (size_t)it * 14;
            const float* ge = genres   + (size_t)it * 18;
            float* dst = &gi[m][96];
            #pragma unroll
            for (int j = 0; j < 14; ++j) dst[j] = ie[j];
            #pragma unroll
            for (int j = 0; j < 18; ++j) dst[14 + j] = ge[j];
        }
    }
    __syncthreads();

    const int row  = lane & 15;              // M for A-fragment
    const int koff = (lane >> 4) << 1;       // 0 (lanes 0-15) or 2 (lanes 16-31)
    const int n    = lane & 15;              // N for B-fragment / C columns

    const float* arow = &gi[row][koff];      // per-lane A base
    const float* bw1  = &sh_aWT[n][koff];    // per-lane B base (GEMM1)
    const float* bw2  = &sh_W1T[n][koff];    // per-lane B base (GEMM2)

    // ---- GEMM1: at_wt[16x3] = gi[16x128] @ attn_W[128x3]  (N padded to 16) ----
    v8f c1 = {};
    #pragma unroll
    for (int s = 0; s < 32; ++s) {
        const int k0 = 4 * s;
        v2f a, b;
        a.x = arow[k0];   a.y = arow[k0 + 1];
        b.x = bw1[k0];    b.y = bw1[k0 + 1];
        c1 = __builtin_amdgcn_wmma_f32_16x16x4_f32(false, a, false, b,
                                                   (short)0, c1, false, false);
    }
    if (n < 3) {   // C layout: vgpr j -> M=j (lanes<16) / M=j+8 (lanes>=16), N=lane%16
        const int mb = (lane < 16) ? 0 : 8;
        const float bia = sh_attnb[n];
        #pragma unroll
        for (int j = 0; j < 8; ++j)
            sh_at[wave][mb + j][n] = c1[j] + bia;
    }
    __syncthreads();

    // ---- g = sum_k at[k]*mem_e[k]; build new = [g*it_e, g, it_e] in place ----
    {
        const int m  = lane & 15;
        const int d0 = (lane >> 4) * 16;
        const float a0 = sh_at[wave][m][0];
        const float a1 = sh_at[wave][m][1];
        const float a2 = sh_at[wave][m][2];
        #pragma unroll
        for (int t = 0; t < 16; ++t) {
            const int d = d0 + t;
            const float gv  = a0 * gi[m][d] + a1 * gi[m][32 + d] + a2 * gi[m][64 + d];
            const float ite = gi[m][96 + d];
            gi[m][d]      = gv * ite;   // g * it_e
            gi[m][32 + d] = gv;         // g
            gi[m][64 + d] = ite;        // it_e
        }
    }
    __syncthreads();

    // ---- GEMM2: h[16x8] = new[16x96] @ pred_W1[96x8]  (N padded to 16) ----
    v8f c2 = {};
    #pragma unroll
    for (int s = 0; s < 24; ++s) {
        const int k0 = 4 * s;
        v2f a, b;
        a.x = arow[k0];   a.y = arow[k0 + 1];
        b.x = bw2[k0];    b.y = bw2[k0 + 1];
        c2 = __builtin_amdgcn_wmma_f32_16x16x4_f32(false, a, false, b,
                                                   (short)0, c2, false, false);
    }
    if (n < 8) {
        const int mb = (lane < 16) ? 0 : 8;
        const float bia = sh_b1[n];
        #pragma unroll
        for (int j = 0; j < 8; ++j) {
            const float hv = c2[j] + bia;
            sh_h[wave][mb + j][n] = hv > 0.0f ? hv : 0.0f;
        }
    }
    __syncthreads();

    // ---- output head: y = sigmoid(h @ W2 + b2) ----
    if (lane < 16) {
        const int m = lane;
        float z = sh_b2;
        #pragma unroll
        for (int q = 0; q < 8; ++q) z += sh_h[wave][m][q] * sh_W2[q];
        const float y = 1.0f / (1.0f + __expf(-z));
        if (s0 + m < B) out[s0 + m] = y;
    }
}

extern "C" void kernel_launch(void* const* d_in, const int* in_sizes, int n_in,
                              void* d_out, int out_size, void* d_ws, size_t ws_size,
                              hipStream_t stream) {
    const int*   group_inputs = (const int*)d_in[0];
    const int*   item_inputs  = (const int*)d_in[1];
    const int*   members      = (const int*)d_in[2];
    const float* user_emb     = (const float*)d_in[3];
    const float* item_emb     = (const float*)d_in[4];
    const float* genres       = (const float*)d_in[5];
    const float* attn_W       = (const float*)d_in[6];
    const float* attn_b       = (const float*)d_in[7];
    const float* pred_W1      = (const float*)d_in[8];
    const float* pred_b1      = (const float*)d_in[9];
    const float* pred_W2      = (const float*)d_in[10];
    const float* pred_b2      = (const float*)d_in[11];

    const int B    = in_sizes[0];
    const int spb  = WAVES * TILE;                 // 64 samples per block
    const int grid = (B + spb - 1) / spb;

    bahdanau_fused_kernel<<<grid, 128, 0, stream>>>(
        group_inputs, item_inputs, members, user_emb, item_emb, genres,
        attn_W, attn_b, pred_W1, pred_b1, pred_W2, pred_b2,
        (float*)d_out, B);
}